// pthdv_block_72516227825926
// MI455X (gfx1250) — compile-verified
//
#include <hip/hip_runtime.h>
#include <hip/hip_bf16.h>
#include <math.h>

typedef float v2f __attribute__((ext_vector_type(2)));
typedef float v8f __attribute__((ext_vector_type(8)));

#define BB 64
#define TT 2048
#define DD 64
#define TM1 2047
#define GDm 16

// workspace layout (float offsets)
#define WS_S  0
#define WS_G  4096
#define WS_PB 8388608
#define WS_SE 8519680

// ---------------- kernel A: build skew-symmetric basis S = tril(W,-1) - tril(W,-1)^T
__global__ void k_init_S(const float* __restrict__ W, float* __restrict__ S) {
  int idx = blockIdx.x * 256 + threadIdx.x;      // 4096 = 4*16*8*8
  int j = idx & 7, i = (idx >> 3) & 7;
  int c = (idx >> 6) & 15, g = idx >> 10;
  float lo = (i > j) ? W[idx] : 0.0f;
  float up = (j > i) ? W[(((g * 16 + c) * 8 + j) * 8) + i] : 0.0f;
  S[idx] = lo - up;
}

// ---------------- kernel B: dX -> Lie algebra -> expm (Taylor12 + 6 squarings) -> LN -> WMMA projection
__global__ __launch_bounds__(64) void k_main(
    const float* __restrict__ x, const float* __restrict__ mask,
    const float* __restrict__ Sg, const float* __restrict__ ln_g,
    const float* __restrict__ ln_b, const float* __restrict__ pw,
    const float* __restrict__ pbias, float* __restrict__ gout) {
  __shared__ float S_lds[16 * 4 * 68];   // [c][g][e], pad 68 avoids bank overlap between g-halves
  __shared__ float f_tile[16 * 260];     // [t_local][256], pad 260 -> conflict-free A-frag reads
  __shared__ float ln_lds[128];
  const int tid = threadIdx.x;
  const int b  = blockIdx.y;
  const int t0 = blockIdx.x * 16;
  const int g  = tid >> 4;               // group constant per wave-half
  const int tl = tid & 15;

  for (int idx = tid; idx < 4096; idx += 64) {
    int e = idx & 63, c = (idx >> 6) & 15, gg = idx >> 10;
    S_lds[(c * 4 + gg) * 68 + e] = Sg[idx];
  }
  for (int idx = tid; idx < 128; idx += 64)
    ln_lds[idx] = (idx < 64) ? ln_g[idx] : ln_b[idx - 64];
  __syncthreads();

  int t  = t0 + tl;
  int tt = (t < TM1) ? t : (TM1 - 1);    // clamp for the partial last tile; store is guarded
  const float* xr = x + ((size_t)b * TT + tt) * DD + g * GDm;
  const float msc = mask[(size_t)b * TT + tt + 1] * 0.015625f;  // fold expm 2^-6 scaling

  // A = sum_c dX_c * S[g][c]   (A pre-scaled by 1/64)
  float A[64];
#pragma unroll
  for (int e = 0; e < 64; ++e) A[e] = 0.0f;
  for (int c = 0; c < 16; ++c) {
    float dxc = (xr[DD + c] - xr[c]) * msc;
    const float4* srow = (const float4*)&S_lds[(c * 4 + g) * 68];
#pragma unroll
    for (int e4 = 0; e4 < 16; ++e4) {
      float4 s4 = srow[e4];
      A[e4 * 4 + 0] = fmaf(dxc, s4.x, A[e4 * 4 + 0]);
      A[e4 * 4 + 1] = fmaf(dxc, s4.y, A[e4 * 4 + 1]);
      A[e4 * 4 + 2] = fmaf(dxc, s4.z, A[e4 * 4 + 2]);
      A[e4 * 4 + 3] = fmaf(dxc, s4.w, A[e4 * 4 + 3]);
    }
  }

  // expm: P = I + sum_{k=1..12} A^k/k!   then 6 squarings, all register-resident
  float P[64], Tm[64];
#pragma unroll
  for (int e = 0; e < 64; ++e) {
    float idv = ((e >> 3) == (e & 7)) ? 1.0f : 0.0f;
    P[e] = idv; Tm[e] = idv;
  }
  for (int k = 1; k <= 12; ++k) {
    float invk = 1.0f / (float)k;
#pragma unroll
    for (int i = 0; i < 8; ++i) {        // new row_i depends only on old row_i -> in-place per row
      float r[8];
#pragma unroll
      for (int j = 0; j < 8; ++j) r[j] = Tm[i * 8] * A[j];
#pragma unroll
      for (int kk = 1; kk < 8; ++kk)
#pragma unroll
        for (int j = 0; j < 8; ++j) r[j] = fmaf(Tm[i * 8 + kk], A[kk * 8 + j], r[j]);
#pragma unroll
      for (int j = 0; j < 8; ++j) {
        float v = r[j] * invk;
        Tm[i * 8 + j] = v;
        P[i * 8 + j] += v;
      }
    }
  }
  for (int sq = 0; sq < 6; ++sq) {
    float Q[64];
#pragma unroll
    for (int i = 0; i < 8; ++i) {
#pragma unroll
      for (int j = 0; j < 8; ++j) Q[i * 8 + j] = P[i * 8] * P[j];
#pragma unroll
      for (int kk = 1; kk < 8; ++kk)
#pragma unroll
        for (int j = 0; j < 8; ++j) Q[i * 8 + j] = fmaf(P[i * 8 + kk], P[kk * 8 + j], Q[i * 8 + j]);
    }
#pragma unroll
    for (int e = 0; e < 64; ++e) P[e] = Q[e];
  }

  // LayerNorm over the 64 matrix entries (per group), write feature tile to LDS
  float s1 = 0.0f, s2 = 0.0f;
#pragma unroll
  for (int e = 0; e < 64; ++e) { s1 += P[e]; s2 += P[e] * P[e]; }
  float mu  = s1 * (1.0f / 64.0f);
  float var = s2 * (1.0f / 64.0f) - mu * mu;
  float rs  = rsqrtf(var + 1e-5f);
  float* frow = &f_tile[tl * 260 + g * 64];
#pragma unroll
  for (int e = 0; e < 64; ++e)
    frow[e] = (P[e] - mu) * rs * ln_lds[e] + ln_lds[64 + e];
  __syncthreads();

  // WMMA projection: (16 x 256) @ (256 x 64) + bias, f32 16x16x4, each wave owns 2 N-tiles
  const int wave  = tid >> 5;
  const int lane  = tid & 31;
  const int row   = lane & 15;
  const int koff  = (lane < 16) ? 0 : 2;   // A 16x4 f32 layout: lanes16-31 hold K=2,3
  const int ncol  = lane & 15;
#pragma unroll
  for (int nt = 0; nt < 2; ++nt) {
    const int n = wave * 32 + nt * 16 + ncol;
    v8f acc;
    float bv = pbias[n];
#pragma unroll
    for (int r = 0; r < 8; ++r) acc[r] = bv;
#pragma unroll 4
    for (int k0 = 0; k0 < 256; k0 += 4) {
      v2f a = *(const v2f*)&f_tile[row * 260 + k0 + koff];
      v2f bfrag;
      bfrag.x = pw[(k0 + koff) * 64 + n];
      bfrag.y = pw[(k0 + koff + 1) * 64 + n];
      acc = __builtin_amdgcn_wmma_f32_16x16x4_f32(false, a, false, bfrag,
                                                  (short)0, acc, false, false);
    }
#pragma unroll
    for (int r = 0; r < 8; ++r) {
      int m = r + ((lane < 16) ? 0 : 8);   // C/D layout: VGPR r -> M=r (lanes<16) / r+8
      int trow = t0 + m;
      if (trow < TM1) gout[((size_t)b * TM1 + trow) * DD + n] = acc[r];
    }
  }
}

// ---------------- kernel C: linear interp 2047->2048, h = x + f, deterministic partial pooling
__global__ void k_interp(const float* __restrict__ x, const float* __restrict__ mask,
                         const float* __restrict__ gbuf, float* __restrict__ hout,
                         float* __restrict__ pb) {
  __shared__ float hbuf[64 * 68];
  const int tid = threadIdx.x;
  const int b = blockIdx.y, chunk = blockIdx.x;
  const int tl = tid >> 2, cg = tid & 3;
  const int t = chunk * 64 + tl;
  float pos = ((float)t + 0.5f) * (2047.0f / 2048.0f) - 0.5f;
  pos = fminf(fmaxf(pos, 0.0f), 2046.0f);
  int i0 = (int)floorf(pos);
  int i1 = (i0 + 1 < TM1) ? i0 + 1 : TM1 - 1;
  float w = pos - (float)i0;
  const float* g0 = gbuf + ((size_t)b * TM1 + i0) * DD + cg * 16;
  const float* g1 = gbuf + ((size_t)b * TM1 + i1) * DD + cg * 16;
  const float* xr = x + ((size_t)b * TT + t) * DD + cg * 16;
  float* orow = hout + ((size_t)b * TT + t) * DD + cg * 16;
  float mv = mask[(size_t)b * TT + t];
  float* hb = &hbuf[tl * 68 + cg * 16];
#pragma unroll
  for (int e = 0; e < 16; ++e) {
    float fi = g0[e] * (1.0f - w) + g1[e] * w;
    float h = xr[e] + fi;
    orow[e] = h;           // stage h in d_out
    hb[e] = h * mv;
  }
  __syncthreads();
  for (int stride = 32; stride >= 1; stride >>= 1) {  // fixed-order tree: deterministic
    if (tl < stride) {
#pragma unroll
      for (int e = 0; e < 16; ++e)
        hbuf[tl * 68 + cg * 16 + e] += hbuf[(tl + stride) * 68 + cg * 16 + e];
    }
    __syncthreads();
  }
  if (tid < 64) pb[((size_t)b * 32 + chunk) * 64 + tid] = hbuf[tid];
}

// ---------------- kernel D: squeeze-excitation gate per batch row
__global__ void k_se(const float* __restrict__ mask, const float* __restrict__ pb,
                     const float* __restrict__ w1, const float* __restrict__ b1,
                     const float* __restrict__ w2, const float* __restrict__ b2,
                     float* __restrict__ sbuf) {
  const int b = threadIdx.x;
  if (b >= BB) return;
  float denom = 0.0f;
  for (int t = 0; t < TT; ++t) denom += mask[(size_t)b * TT + t];
  float pooled[64];
#pragma unroll
  for (int e = 0; e < 64; ++e) pooled[e] = 0.0f;
  for (int c = 0; c < 32; ++c) {
    const float* row = pb + ((size_t)b * 32 + c) * 64;
#pragma unroll
    for (int e = 0; e < 64; ++e) pooled[e] += row[e];
  }
  float inv = 1.0f / denom;
#pragma unroll
  for (int e = 0; e < 64; ++e) pooled[e] *= inv;
  float hid[4];
#pragma unroll
  for (int j = 0; j < 4; ++j) {
    float a = b1[j];
#pragma unroll
    for (int e = 0; e < 64; ++e) a = fmaf(pooled[e], w1[e * 4 + j], a);
    hid[j] = 0.5f * a * (1.0f + erff(a * 0.70710678118654752f));  // exact gelu
  }
  for (int n = 0; n < 64; ++n) {
    float a = b2[n];
#pragma unroll
    for (int j = 0; j < 4; ++j) a = fmaf(hid[j], w2[j * 64 + n], a);
    sbuf[(size_t)b * 64 + n] = 1.0f / (1.0f + expf(-a));
  }
}

// ---------------- kernel E: out = h * s + x (in place over staged h in d_out)
__global__ void k_final(const float* __restrict__ x, const float* __restrict__ sbuf,
                        float* __restrict__ out) {
  size_t f4 = (size_t)blockIdx.x * 256 + threadIdx.x;  // float4 index, 2^21 total
  size_t b  = f4 >> 15;                                 // T*D/4 = 32768
  int ch    = ((int)(f4 & 15)) * 4;
  float4 h4 = ((const float4*)out)[f4];
  float4 x4 = ((const float4*)x)[f4];
  float4 sv = *(const float4*)&sbuf[b * 64 + ch];
  float4 r;
  r.x = fmaf(h4.x, sv.x, x4.x);
  r.y = fmaf(h4.y, sv.y, x4.y);
  r.z = fmaf(h4.z, sv.z, x4.z);
  r.w = fmaf(h4.w, sv.w, x4.w);
  ((float4*)out)[f4] = r;
}

// ---------------- kernel F: mask passthrough (second tuple output)
__global__ void k_maskcopy(const float* __restrict__ mask, float* __restrict__ mo) {
  int idx = blockIdx.x * 256 + threadIdx.x;  // 131072
  mo[idx] = mask[idx];
}

extern "C" void kernel_launch(void* const* d_in, const int* in_sizes, int n_in,
                              void* d_out, int out_size, void* d_ws, size_t ws_size,
                              hipStream_t stream) {
  const float* x      = (const float*)d_in[0];
  const float* mask   = (const float*)d_in[1];
  const float* W_dev  = (const float*)d_in[2];
  const float* ln_g   = (const float*)d_in[3];
  const float* ln_b   = (const float*)d_in[4];
  const float* proj_w = (const float*)d_in[5];
  const float* proj_b = (const float*)d_in[6];
  const float* se_w1  = (const float*)d_in[7];
  const float* se_b1  = (const float*)d_in[8];
  const float* se_w2  = (const float*)d_in[9];
  const float* se_b2  = (const float*)d_in[10];
  float* out = (float*)d_out;
  float* ws  = (float*)d_ws;
  float* S    = ws + WS_S;
  float* gbuf = ws + WS_G;
  float* pb   = ws + WS_PB;
  float* sbuf = ws + WS_SE;

  k_init_S<<<16, 256, 0, stream>>>(W_dev, S);
  k_main<<<dim3(128, BB), 64, 0, stream>>>(x, mask, S, ln_g, ln_b, proj_w, proj_b, gbuf);
  k_interp<<<dim3(32, BB), 256, 0, stream>>>(x, mask, gbuf, out, pb);
  k_se<<<1, 64, 0, stream>>>(mask, pb, se_w1, se_b1, se_w2, se_b2, sbuf);
  k_final<<<8192, 256, 0, stream>>>(x, sbuf, out);
  k_maskcopy<<<512, 256, 0, stream>>>(mask, out + (size_t)BB * TT * DD);
}